// BPF_10986526343936
// MI455X (gfx1250) — compile-verified
//
#include <hip/hip_runtime.h>
#include <hip/hip_bf16.h>
#include <math.h>

#define NXI 256
#define NPIX (256 * 256)
#define NSAMP 128
#define BROWS 32   // A-panel rows staged in LDS per GEMM block

typedef __attribute__((ext_vector_type(2))) float v2f;
typedef __attribute__((ext_vector_type(8))) float v8f;

// ---------------- i0e (exponentially-scaled modified Bessel I0), A&S 9.8 ----
__device__ __forceinline__ float i0e_dev(float x) {
    // x >= 0 here
    if (x < 3.75f) {
        float t = x * (1.0f / 3.75f);
        t = t * t;
        float p = 1.0f + t * (3.5156229f + t * (3.0899424f + t * (1.2067492f +
                  t * (0.2659732f + t * (0.0360768f + t * 0.0045813f)))));
        return p * __expf(-x);
    } else {
        float t = 3.75f / x;
        float p = 0.39894228f + t * (0.01328592f + t * (0.00225319f + t * (-0.00157565f +
                  t * (0.00916281f + t * (-0.02057706f + t * (0.02635537f +
                  t * (-0.01647633f + t * 0.00392377f)))))));
        return p * rsqrtf(x);
    }
}

// ---------------- init: zero accumulator, DFT twiddles, fused shift+filter ---
__global__ __launch_bounds__(256) void init_kernel(float* __restrict__ img,
                                                   float* __restrict__ Wr,
                                                   float* __restrict__ Wi,
                                                   float* __restrict__ H) {
    int idx = blockIdx.x * blockDim.x + threadIdx.x;   // 0..65535
    img[idx] = 0.0f;

    int j = idx >> 8;
    int k = idx & 255;

    // W[j,k] = exp(-2*pi*i*j*k/256); exact phase reduction mod 256
    int ph = (j * k) & 255;
    float ang = -6.28318530717958647692f * (float)ph * (1.0f / 256.0f);
    float s, c;
    sincosf(ang, &s, &c);
    Wr[idx] = c;
    Wi[idx] = s;

    // H = ifftshift(freq_filter) * NX, with freq_filter = 1/i0e(w0*(pi*ts)^2)
    int js = (j + 128) & 255;
    int ks = (k + 128) & 255;
    float xs = ((float)js - 127.5f) * (1.0f / 128.0f);
    float ys = ((float)ks - 127.5f) * (1.0f / 128.0f);
    // tof_sigma_pixels = 300*0.3/2/2.355/2 = 9.554140127...
    const float PTS2 = (float)(M_PI * M_PI * 9.554140127388535 * 9.554140127388535);
    float tmp = (xs * xs + ys * ys) * PTS2;
    H[idx] = 256.0f / i0e_dev(tmp);
}

// ---------------- TOF backprojection with LDS-privatized image --------------
__global__ __launch_bounds__(256) void bp_kernel(
    const float* __restrict__ proj, const float* __restrict__ tof,
    const float* __restrict__ x1l, const float* __restrict__ y1l,
    const float* __restrict__ x1r, const float* __restrict__ y1r,
    const float* __restrict__ x2l, const float* __restrict__ y2l,
    const float* __restrict__ x2r, const float* __restrict__ y2r,
    float* __restrict__ img, int E) {
    extern __shared__ float limg[];   // full 256x256 image: 256KB of the 320KB WGP LDS
    for (int i = threadIdx.x; i < NPIX; i += blockDim.x) limg[i] = 0.0f;
    __syncthreads();

    const float inv_sigma = 1.0f / 19.108280254777070f;  // sigma = 300*0.3/2/2.355 mm
    int stride = gridDim.x * blockDim.x;
    for (int e = blockIdx.x * blockDim.x + threadIdx.x; e < E; e += stride) {
        // prefetch next grid-stride iteration's event data (global_prefetch_b8)
        int ep = (e + stride < E) ? (e + stride) : e;
        __builtin_prefetch(&proj[ep], 0, 0);
        __builtin_prefetch(&x1l[ep], 0, 0);
        __builtin_prefetch(&x2l[ep], 0, 0);

        float X1 = 0.5f * (x1l[e] + x1r[e]);
        float Y1 = 0.5f * (y1l[e] + y1r[e]);
        float X2 = 0.5f * (x2l[e] + x2r[e]);
        float Y2 = 0.5f * (y2l[e] + y2r[e]);
        float dx = X2 - X1, dy = Y2 - Y1;
        float L = sqrtf(dx * dx + dy * dy);
        float center = 0.5f * L + tof[e] * 0.15f;
        float step = L * (1.0f / (float)NSAMP);
        float pw = proj[e] * step;
        #pragma unroll 4
        for (int i = 0; i < NSAMP; ++i) {
            float t = ((float)i + 0.5f) * (1.0f / (float)NSAMP);
            float z = (t * L - center) * inv_sigma;
            float w = __expf(-0.5f * z * z) * pw;
            float px = X1 + t * dx;
            float py = Y1 + t * dy;
            int ix = (int)floorf(px * 0.5f + 128.0f);
            int iy = (int)floorf(py * 0.5f + 128.0f);
            if (((unsigned)ix < 256u) && ((unsigned)iy < 256u)) {
                atomicAdd(&limg[(iy << 8) + ix], w);   // ds_add_f32
            }
        }
    }
    __syncthreads();
    for (int i = threadIdx.x; i < NPIX; i += blockDim.x) {
        float v = limg[i];
        if (v != 0.0f) atomicAdd(&img[i], v);          // global_atomic_add_f32
    }
}

// ---------------- f32 WMMA GEMM: D = alpha*(A x B) + beta*D, 256x256x256 ----
// Block = 8 waves covering a 32x256 output panel. The block's 32x256 A panel
// (contiguous 32KB) is staged into LDS with gfx1250 async copies; each wave
// computes a 16x64 strip = 4 tiles sharing one A fragment per K-step.
__global__ __launch_bounds__(256) void wmma_gemm_256(
    const float* __restrict__ A, const float* __restrict__ B,
    float* __restrict__ D, float alpha, float beta) {
    __shared__ __align__(16) float sA[BROWS * NXI];    // 32KB
    int tid = threadIdx.x;
    int rowbase = blockIdx.x * BROWS;

    // ---- async stage of the A panel: GLOBAL_LOAD_ASYNC_TO_LDS_B128 ----
    const float* gA = A + rowbase * NXI;               // contiguous 32KB chunk
    #pragma unroll
    for (int c = 0; c < 8; ++c) {
        int elt = (c * 256 + tid) * 4;                 // float index, 16B granules
        unsigned ldsOff = (unsigned)(unsigned long long)(const void*)&sA[elt];
        unsigned long long g = (unsigned long long)(const void*)(gA + elt);
        asm volatile("global_load_async_to_lds_b128 %0, %1, off"
                     :: "v"(ldsOff), "v"(g) : "memory");
    }
    asm volatile("s_wait_asynccnt 0x0" ::: "memory");
    __syncthreads();

    int wave = tid >> 5;
    int lane = tid & 31;
    int rg = wave >> 2;            // row group within block (0..1)
    int cg = wave & 3;             // col group (0..3), 64 columns each
    int hl = lane >> 4;            // half-wave select
    int r  = lane & 15;
    int tm  = rowbase + rg * 16;   // output row base for this wave
    int tn0 = cg * 64;             // output col base for this wave

    v8f acc[4] = {{}, {}, {}, {}};
    const float* sArow = &sA[(rg * 16 + r) * NXI];
    for (int k = 0; k < NXI; k += 4) {
        // A fragment (16x4) from LDS: v0 = K0|K2, v1 = K1|K3 across half-waves
        v2f a;
        a.x = sArow[k + 2 * hl + 0];
        a.y = sArow[k + 2 * hl + 1];
        // Four B fragments (4x16 each), rows striped across lanes
        const float* Brow0 = B + (k + 2 * hl) * NXI + tn0 + r;
        const float* Brow1 = Brow0 + NXI;
        #pragma unroll
        for (int t = 0; t < 4; ++t) {
            v2f b;
            b.x = Brow0[t * 16];
            b.y = Brow1[t * 16];
            acc[t] = __builtin_amdgcn_wmma_f32_16x16x4_f32(
                /*neg_a=*/false, a, /*neg_b=*/false, b,
                /*c_mod=*/(short)0, acc[t], /*reuse_a=*/false, /*reuse_b=*/false);
        }
    }
    #pragma unroll
    for (int t = 0; t < 4; ++t) {
        #pragma unroll
        for (int v = 0; v < 8; ++v) {
            int row = tm + v + 8 * hl;     // VGPR v holds rows v / v+8 per half-wave
            int idx = row * NXI + tn0 + t * 16 + r;
            float out = alpha * acc[t][v];
            if (beta != 0.0f) out += beta * D[idx];
            D[idx] = out;
        }
    }
}

// ---------------- pointwise: apply frequency filter to complex spectrum -----
__global__ __launch_bounds__(256) void filt_kernel(float* __restrict__ Fr,
                                                   float* __restrict__ Fi,
                                                   const float* __restrict__ H) {
    int idx = blockIdx.x * blockDim.x + threadIdx.x;
    float h = H[idx];
    Fr[idx] *= h;
    Fi[idx] *= h;
}

extern "C" void kernel_launch(void* const* d_in, const int* in_sizes, int n_in,
                              void* d_out, int out_size, void* d_ws, size_t ws_size,
                              hipStream_t stream) {
    const float* proj = (const float*)d_in[0];
    const float* tof  = (const float*)d_in[1];
    const float* x1l  = (const float*)d_in[2];
    const float* y1l  = (const float*)d_in[3];
    const float* x1r  = (const float*)d_in[4];
    const float* y1r  = (const float*)d_in[5];
    const float* x2l  = (const float*)d_in[6];
    const float* y2l  = (const float*)d_in[7];
    const float* x2r  = (const float*)d_in[8];
    const float* y2r  = (const float*)d_in[9];
    int E = in_sizes[0];

    float* w   = (float*)d_ws;
    float* img = w + 0 * NPIX;   // backprojection accumulator (= G * 4)
    float* Wr  = w + 1 * NPIX;   // DFT twiddle real
    float* Wi  = w + 2 * NPIX;   // DFT twiddle imag
    float* H   = w + 3 * NPIX;   // fused ifftshift(filter)*NX
    float* Tr  = w + 4 * NPIX;
    float* Ti  = w + 5 * NPIX;
    float* Fr  = w + 6 * NPIX;
    float* Fi  = w + 7 * NPIX;
    float* Pr  = w + 8 * NPIX;
    float* Pi  = w + 9 * NPIX;
    float* out = (float*)d_out;

    const int GB = NXI / BROWS;  // GEMM grid: 8 blocks x 256 threads

    init_kernel<<<NPIX / 256, 256, 0, stream>>>(img, Wr, Wi, H);
    bp_kernel<<<128, 256, NPIX * sizeof(float), stream>>>(
        proj, tof, x1l, y1l, x1r, y1r, x2l, y2l, x2r, y2r, img, E);

    // G = img/4 (folded into alpha). T = G * W
    wmma_gemm_256<<<GB, 256, 0, stream>>>(img, Wr, Tr, 0.25f, 0.0f);
    wmma_gemm_256<<<GB, 256, 0, stream>>>(img, Wi, Ti, 0.25f, 0.0f);
    // F = W * T  (full 2D forward DFT)
    wmma_gemm_256<<<GB, 256, 0, stream>>>(Wr, Tr, Fr, 1.0f, 0.0f);
    wmma_gemm_256<<<GB, 256, 0, stream>>>(Wi, Ti, Fr, -1.0f, 1.0f);
    wmma_gemm_256<<<GB, 256, 0, stream>>>(Wr, Ti, Fi, 1.0f, 0.0f);
    wmma_gemm_256<<<GB, 256, 0, stream>>>(Wi, Tr, Fi, 1.0f, 1.0f);
    // apply TOF de-blurring filter (shifts folded in)
    filt_kernel<<<NPIX / 256, 256, 0, stream>>>(Fr, Fi, H);
    // P = F * conj(W)
    wmma_gemm_256<<<GB, 256, 0, stream>>>(Fr, Wr, Pr, 1.0f, 0.0f);
    wmma_gemm_256<<<GB, 256, 0, stream>>>(Fi, Wi, Pr, 1.0f, 1.0f);
    wmma_gemm_256<<<GB, 256, 0, stream>>>(Fi, Wr, Pi, 1.0f, 0.0f);
    wmma_gemm_256<<<GB, 256, 0, stream>>>(Fr, Wi, Pi, -1.0f, 1.0f);
    // image = Re[conj(W) * P] / N^2
    wmma_gemm_256<<<GB, 256, 0, stream>>>(Wr, Pr, out, 1.0f / 65536.0f, 0.0f);
    wmma_gemm_256<<<GB, 256, 0, stream>>>(Wi, Pi, out, 1.0f / 65536.0f, 1.0f);

    (void)n_in; (void)out_size; (void)ws_size;
}